// RNN_MultiRegional_D1D2_61091614818738
// MI455X (gfx1250) — compile-verified
//
#include <hip/hip_runtime.h>

// ---------------- Problem constants ----------------
#define HID     520
#define FSI_N   156
#define INP_N   4
#define HALF    260
#define TOTAL   3800
#define BATCH   8
#define T_STEPS 1000
#define ALPHA   0.1f

#define I_D1    0
#define I_D2    520
#define I_FSI   1040
#define I_GPE   1196
#define I_STN   1716
#define I_SNR   2236
#define I_THAL  2756
#define I_ALM   3276
#define I_ITI   3796

#define NTT     238          // output (N) tiles of 16 -> 3808 padded
#define NKT     950          // K tiles of 4 -> 3800 exact
#define LDH     3808         // padded h row stride (floats)
#define NWAVES  4

typedef __attribute__((ext_vector_type(2))) float v2f;
typedef __attribute__((ext_vector_type(8))) float v8f;

struct WParams {
    const float* d12d1;  const float* d22d2;  const float* d12d2;  const float* d22d1;
    const float* thal2alm; const float* thal2d1; const float* thal2d2;
    const float* alm2alm; const float* alm2d1; const float* alm2d2;
    const float* d12snr; const float* d22gpe; const float* gpe2stn;
    const float* stn2snr; const float* snr2thal;
    const float* fsi2d1; const float* fsi2d2; const float* thal2fsi;
    const float* alm2fsi; const float* iti2fsi; const float* fsi2fsi;
    const int*   smask;
};

__device__ __forceinline__ float reluf(float x) { return x > 0.0f ? x : 0.0f; }

// W[n][k] per the reference block assembly.
__device__ float wval(int n, int k, const WParams& P) {
    if (n < I_D2) {                       // D1 rows
        int i = n;
        if (k < I_D2)  { int j = k;         float m = P.smask[i*HID+j] ? 1.f : 0.f; return -m * reluf(P.d12d1[i*HID+j]); }
        if (k < I_FSI) { int j = k - I_D2;  float m = P.smask[i*HID+j] ? 1.f : 0.f; return -m * reluf(P.d22d1[i*HID+j]); }
        if (k < I_GPE) { int j = k - I_FSI; return -reluf(P.fsi2d1[i*FSI_N+j]); }
        if (k >= I_THAL && k < I_ALM) return reluf(P.thal2d1[i*HID + (k - I_THAL)]);
        if (k >= I_ALM  && k < I_ITI) { int j = k - I_ALM; return (j < HID - FSI_N) ? reluf(P.alm2d1[i*HID+j]) : 0.f; }
        return 0.f;
    }
    if (n < I_FSI) {                      // D2 rows
        int i = n - I_D2;
        if (k < I_D2)  { int j = k;         float m = P.smask[i*HID+j] ? 1.f : 0.f; return -m * reluf(P.d12d2[i*HID+j]); }
        if (k < I_FSI) { int j = k - I_D2;  float m = P.smask[i*HID+j] ? 1.f : 0.f; return -m * reluf(P.d22d2[i*HID+j]); }
        if (k < I_GPE) { int j = k - I_FSI; return -reluf(P.fsi2d2[i*FSI_N+j]); }
        if (k >= I_THAL && k < I_ALM) return reluf(P.thal2d2[i*HID + (k - I_THAL)]);
        if (k >= I_ALM  && k < I_ITI) { int j = k - I_ALM; return (j < HID - FSI_N) ? reluf(P.alm2d2[i*HID+j]) : 0.f; }
        return 0.f;
    }
    if (n < I_GPE) {                      // FSI rows
        int i = n - I_FSI;
        if (k >= I_FSI  && k < I_GPE) return -reluf(P.fsi2fsi[i*FSI_N + (k - I_FSI)]);
        if (k >= I_THAL && k < I_ALM) return reluf(P.thal2fsi[i*HID + (k - I_THAL)]);
        if (k >= I_ALM  && k < I_ITI) { int j = k - I_ALM; return (j < HID - FSI_N) ? reluf(P.alm2fsi[i*HID+j]) : 0.f; }
        if (k >= I_ITI  && k < TOTAL) return reluf(P.iti2fsi[i*INP_N + (k - I_ITI)]);
        return 0.f;
    }
    if (n < I_STN) {                      // GPE rows
        int i = n - I_GPE;
        if (k >= I_D2 && k < I_FSI) { int j = k - I_D2; float bd = ((i < HALF) == (j < HALF)) ? 1.f : 0.f; return -bd * reluf(P.d22gpe[i*HID+j]); }
        return 0.f;
    }
    if (n < I_SNR) {                      // STN rows
        int i = n - I_STN;
        if (k >= I_GPE && k < I_STN) { int j = k - I_GPE; float bd = ((i < HALF) == (j < HALF)) ? 1.f : 0.f; return -bd * reluf(P.gpe2stn[i*HID+j]); }
        return 0.f;
    }
    if (n < I_THAL) {                     // SNR rows
        int i = n - I_SNR;
        if (k >= I_STN && k < I_SNR) { int j = k - I_STN; float bd = ((i < HALF) == (j < HALF)) ? 1.f : 0.f; return  bd * reluf(P.stn2snr[i*HID+j]); }
        if (k < I_D2)                { int j = k;         float bd = ((i < HALF) == (j < HALF)) ? 1.f : 0.f; return -bd * reluf(P.d12snr[i*HID+j]); }
        return 0.f;
    }
    if (n < I_ALM) {                      // THAL rows
        int i = n - I_THAL;
        if (k >= I_SNR && k < I_THAL) { int j = k - I_SNR; float bd = ((i < HALF) == (j < HALF)) ? 1.f : 0.f; return -bd * reluf(P.snr2thal[i*HID+j]); }
        return 0.f;
    }
    if (n < I_ITI) {                      // ALM rows
        int i = n - I_ALM;
        if (k >= I_THAL && k < I_ALM) return reluf(P.thal2alm[i*HID + (k - I_THAL)]);
        if (k >= I_ALM  && k < I_ITI) { int j = k - I_ALM; float dale = (j < HID - FSI_N) ? 1.f : -1.f; return dale * reluf(P.alm2alm[i*HID+j]); }
        return 0.f;
    }
    return 0.f;                           // ITI rows: no recurrent input
}

// Build Wᵀ pre-swizzled into WMMA B-fragment layout, K-contiguous per n-tile:
// element idx = ((nt*NKT + kt)*32 + lane)*2 + v  holds  W[n][k]
// with k = 4*kt + 2*(lane/16) + v , n = 16*nt + lane%16.
__global__ __launch_bounds__(256) void build_wt_kernel(WParams P, float* __restrict__ WTs) {
    size_t idx = (size_t)blockIdx.x * blockDim.x + threadIdx.x;
    const size_t NELEM = (size_t)NKT * NTT * 64;
    if (idx >= NELEM) return;
    int v    = (int)(idx & 1);
    int lane = (int)((idx >> 1) & 31);
    size_t tile = idx >> 6;
    int kt = (int)(tile % NKT);
    int nt = (int)(tile / NKT);
    int k = kt * 4 + ((lane >> 4) << 1) + v;
    int n = nt * 16 + (lane & 15);
    float w = (n < TOTAL) ? wval(n, k, P) : 0.0f;
    WTs[idx] = w;
}

// Initialize both padded h buffers: rows 0..7 of h0 = hn, everything else 0.
__global__ __launch_bounds__(256) void init_h_kernel(const float* __restrict__ hn,
                                                     float* __restrict__ h0,
                                                     float* __restrict__ h1) {
    int idx = blockIdx.x * blockDim.x + threadIdx.x;
    const int PER = 16 * LDH;
    if (idx >= 2 * PER) return;
    float* dst = (idx < PER) ? h0 : h1;
    int o = (idx < PER) ? idx : idx - PER;
    int m = o / LDH, col = o % LDH;
    float val = 0.0f;
    if (idx < PER && m < BATCH && col < TOTAL) val = hn[m * TOTAL + col];
    dst[o] = val;
}

__device__ __forceinline__ float tonic_of(int n) {
    if (n < I_GPE)  return 0.01f;   // d1, d2, fsi
    if (n < I_STN)  return 0.80f;   // gpe
    if (n < I_SNR)  return 0.60f;   // stn
    if (n < I_THAL) return 0.80f;   // snr
    if (n < I_ITI)  return 0.01f;   // thal, alm
    return 0.0f;                    // iti
}

// One Euler step: one 16-output tile per block; f32 WMMA over nonzero K regions.
__global__ __launch_bounds__(128) void step_kernel(const float* __restrict__ WTs,
                                                   const float* __restrict__ hcur,
                                                   float* __restrict__ hnext,
                                                   const float* __restrict__ noise,
                                                   const float* __restrict__ inp,
                                                   float* __restrict__ out,
                                                   int t) {
    const int RS[9] = {I_D1, I_D2, I_FSI, I_GPE, I_STN, I_SNR, I_THAL, I_ALM, I_ITI};
    const int RE[9] = {I_D2, I_FSI, I_GPE, I_STN, I_SNR, I_THAL, I_ALM, I_ITI, TOTAL};
    // per-output-region mask of source regions (bit r = source region r)
    const unsigned OM[9] = {0x0C7u, 0x0C7u, 0x1C4u, 0x002u, 0x008u, 0x011u, 0x020u, 0x0C0u, 0x000u};

    const int nt   = blockIdx.x;
    const int lane = (int)(threadIdx.x & 31);
    // scalarize the wave id so per-wave loop bounds are SGPR-uniform
    const int wave = __builtin_amdgcn_readfirstlane((int)(threadIdx.x >> 5));
    const int n0   = nt * 16;

    unsigned mask = 0;
    #pragma unroll
    for (int r = 0; r < 9; ++r)
        if (n0 < RE[r] && n0 + 16 > RS[r]) mask |= OM[r];

    const int m    = lane & 15;
    const int half = lane >> 4;
    const float* arow  = hcur + (size_t)m * LDH + 2 * half;       // A frag: k = 4*kt + 2*half + {0,1}
    const float* bbase = WTs + (size_t)nt * NKT * 64 + lane * 2;  // B frag stride: 64 floats per kt

    v8f c0 = {}, c1 = {};
    for (int r = 0; r < 9; ++r) {
        if (!((mask >> r) & 1u)) continue;
        const int kt0 = RS[r] >> 2, kt1 = RE[r] >> 2;
        const int len = kt1 - kt0;
        const int q = len >> 2, rem = len & 3;
        int kb = kt0 + wave * q + (wave < rem ? wave : rem);   // contiguous chunk per wave
        int ke = kb + q + (wave < rem ? 1 : 0);
        const float* bp = bbase + (size_t)kb * 64;
        int kt = kb;
        #pragma unroll 2
        for (; kt + 1 < ke; kt += 2) {
            __builtin_prefetch(bp + 8 * 64, 0, 3);             // ~2KB ahead in the B stream
            v2f a0 = *(const v2f*)(arow + kt * 4);
            v2f b0 = *(const v2f*)(bp);
            v2f a1 = *(const v2f*)(arow + kt * 4 + 4);
            v2f b1 = *(const v2f*)(bp + 64);
            c0 = __builtin_amdgcn_wmma_f32_16x16x4_f32(false, a0, false, b0, (short)0, c0, false, false);
            c1 = __builtin_amdgcn_wmma_f32_16x16x4_f32(false, a1, false, b1, (short)0, c1, false, false);
            bp += 128;
        }
        if (kt < ke) {
            v2f a0 = *(const v2f*)(arow + kt * 4);
            v2f b0 = *(const v2f*)(bp);
            c0 = __builtin_amdgcn_wmma_f32_16x16x4_f32(false, a0, false, b0, (short)0, c0, false, false);
        }
    }
    v8f c = c0 + c1;

    __shared__ float red[NWAVES][32][8];
    #pragma unroll
    for (int i = 0; i < 8; ++i) red[wave][lane][i] = c[i];
    __syncthreads();

    if (wave == 0 && lane < 16) {
        int n = n0 + lane;
        if (n < TOTAL) {
            float ton = tonic_of(n);
            #pragma unroll
            for (int b = 0; b < BATCH; ++b) {
                // D row M=b lives in VGPR b for lanes 0..15
                float acc = red[0][lane][b] + red[1][lane][b] + red[2][lane][b] + red[3][lane][b];
                float ext = ton + noise[((size_t)b * T_STEPS + t) * TOTAL + n];
                if (n >= I_ITI) ext += inp[((size_t)b * T_STEPS + t) * INP_N + (n - I_ITI)];
                float h  = hcur[(size_t)b * LDH + n];
                float hv = (1.0f - ALPHA) * h + ALPHA * (acc + ext);
                hv = hv > 0.0f ? hv : 0.0f;
                hnext[(size_t)b * LDH + n] = hv;
                out[((size_t)b * T_STEPS + t) * TOTAL + n] = hv;
            }
        }
    }
}

extern "C" void kernel_launch(void* const* d_in, const int* in_sizes, int n_in,
                              void* d_out, int out_size, void* d_ws, size_t ws_size,
                              hipStream_t stream) {
    WParams P;
    P.d12d1    = (const float*)d_in[0];
    P.d22d2    = (const float*)d_in[1];
    P.d12d2    = (const float*)d_in[2];
    P.d22d1    = (const float*)d_in[3];
    P.thal2alm = (const float*)d_in[4];
    P.thal2d1  = (const float*)d_in[5];
    P.thal2d2  = (const float*)d_in[6];
    P.alm2alm  = (const float*)d_in[7];
    P.alm2d1   = (const float*)d_in[8];
    P.alm2d2   = (const float*)d_in[9];
    P.d12snr   = (const float*)d_in[10];
    P.d22gpe   = (const float*)d_in[11];
    P.gpe2stn  = (const float*)d_in[12];
    P.stn2snr  = (const float*)d_in[13];
    P.snr2thal = (const float*)d_in[14];
    P.fsi2d1   = (const float*)d_in[15];
    P.fsi2d2   = (const float*)d_in[16];
    P.thal2fsi = (const float*)d_in[17];
    P.alm2fsi  = (const float*)d_in[18];
    P.iti2fsi  = (const float*)d_in[19];
    P.fsi2fsi  = (const float*)d_in[20];
    P.smask    = (const int*)d_in[21];
    const float* inp   = (const float*)d_in[22];
    const float* hn    = (const float*)d_in[23];
    const float* noise = (const float*)d_in[24];
    float* out = (float*)d_out;

    const size_t WTS_FLOATS  = (size_t)NKT * NTT * 64;   // 14,470,400 floats (~57.9 MB)
    const size_t HBUF_FLOATS = (size_t)16 * LDH;
    float* WTs = (float*)d_ws;
    float* h0  = WTs + WTS_FLOATS;
    float* h1  = h0 + HBUF_FLOATS;

    {   // Assemble pre-swizzled Wᵀ (L2-resident for the whole step loop)
        size_t nelem = WTS_FLOATS;
        int blocks = (int)((nelem + 255) / 256);
        build_wt_kernel<<<blocks, 256, 0, stream>>>(P, WTs);
    }
    {   // Init padded h double-buffers
        int nelem = 2 * 16 * LDH;
        int blocks = (nelem + 255) / 256;
        init_h_kernel<<<blocks, 256, 0, stream>>>(hn, h0, h1);
    }
    for (int t = 0; t < T_STEPS; ++t) {
        const float* hc = (t & 1) ? h1 : h0;
        float*       hx = (t & 1) ? h0 : h1;
        step_kernel<<<NTT, 128, 0, stream>>>(WTs, hc, hx, noise, inp, out, t);
    }
}